// LNN_36386962931981
// MI455X (gfx1250) — compile-verified
//
#include <hip/hip_runtime.h>
#include <cmath>

// ---------------------------------------------------------------------------
// Problem constants (match reference): N=1024, HID=4096, z-dim = 2N = 2048
// ---------------------------------------------------------------------------
#define NDIM 1024
#define ZDIM 2048
#define HID  4096
#define NS_ITERS 36

typedef __attribute__((ext_vector_type(2))) float v2f;
typedef __attribute__((ext_vector_type(8))) float v8f;

__device__ __forceinline__ float sigmoidf_(float z) {
    return 1.0f / (1.0f + expf(-z));
}
__device__ __forceinline__ float softplusf_(float z) {
    // stable: max(z,0) + log1p(exp(-|z|))
    return fmaxf(z, 0.0f) + log1pf(expf(-fabsf(z)));
}

// ---------------------------------------------------------------------------
// layer1: a1 = x @ W1 + b1 ; s1 = sigmoid(a1), h1 = softplus(a1)
// one thread per output column, x staged in LDS. grid = HID/256
// ---------------------------------------------------------------------------
__global__ __launch_bounds__(256) void layer1_kernel(
    const float* __restrict__ x, const float* __restrict__ W1,
    const float* __restrict__ b1,
    float* __restrict__ s1, float* __restrict__ h1)
{
    __shared__ float xs[ZDIM];
    for (int i = threadIdx.x; i < ZDIM; i += 256) xs[i] = x[i];
    __syncthreads();
    const int j = blockIdx.x * 256 + threadIdx.x;
    float acc = b1[j];
    for (int i = 0; i < ZDIM; ++i)
        acc += xs[i] * W1[(size_t)i * HID + j];
    s1[j] = sigmoidf_(acc);
    h1[j] = softplusf_(acc);
}

// ---------------------------------------------------------------------------
// layer2: a2 = h1 @ W2 + b2 ; g2 = s2*W3 ; d2 = s2*(1-s2)*W3
// ---------------------------------------------------------------------------
__global__ __launch_bounds__(256) void layer2_kernel(
    const float* __restrict__ h1, const float* __restrict__ W2,
    const float* __restrict__ b2, const float* __restrict__ W3,
    float* __restrict__ g2, float* __restrict__ d2)
{
    __shared__ float hs[HID];
    for (int i = threadIdx.x; i < HID; i += 256) hs[i] = h1[i];
    __syncthreads();
    const int j = blockIdx.x * 256 + threadIdx.x;
    float acc = b2[j];
    for (int i = 0; i < HID; ++i)
        acc += hs[i] * W2[(size_t)i * HID + j];
    const float s  = sigmoidf_(acc);
    const float w3 = W3[j];
    g2[j] = s * w3;
    d2[j] = s * (1.0f - s) * w3;
}

// ---------------------------------------------------------------------------
// u_k = sum_j W2[k,j]*g2[j] ; g1 = s1*u ; d1 = s1*(1-s1)*u
// one wave32 per row.  grid = HID/8, block = 256 (8 waves)
// ---------------------------------------------------------------------------
__global__ __launch_bounds__(256) void row_u_kernel(
    const float* __restrict__ W2, const float* __restrict__ g2,
    const float* __restrict__ s1,
    float* __restrict__ g1, float* __restrict__ d1)
{
    const int lane = threadIdx.x & 31;
    const int row  = blockIdx.x * 8 + (threadIdx.x >> 5);
    const float* wr = W2 + (size_t)row * HID;
    float acc = 0.0f;
    for (int j = lane; j < HID; j += 32) acc += wr[j] * g2[j];
#pragma unroll
    for (int off = 16; off > 0; off >>= 1) acc += __shfl_xor(acc, off);
    if (lane == 0) {
        const float s = s1[row];
        g1[row] = s * acc;
        d1[row] = s * (1.0f - s) * acc;
    }
}

// ---------------------------------------------------------------------------
// generic row matvec:  y[row] = alpha * (W[row,:] . v) + (addv ? addv[row] : 0)
// one wave32 per row.  grid = nrows/8, block = 256
// ---------------------------------------------------------------------------
__global__ __launch_bounds__(256) void row_matvec(
    const float* __restrict__ W, int ld, int ncols,
    const float* __restrict__ v,
    const float* __restrict__ addv, float alpha,
    float* __restrict__ y, int nrows)
{
    const int lane = threadIdx.x & 31;
    const int row  = blockIdx.x * 8 + (threadIdx.x >> 5);
    if (row >= nrows) return;
    const float* wr = W + (size_t)row * ld;
    float acc = 0.0f;
    for (int j = lane; j < ncols; j += 32) acc += wr[j] * v[j];
#pragma unroll
    for (int off = 16; off > 0; off >>= 1) acc += __shfl_xor(acc, off);
    if (lane == 0)
        y[row] = alpha * acc + (addv ? addv[row] : 0.0f);
}

// ---------------------------------------------------------------------------
// Tiled fp32 WMMA GEMM:   C(MxN) = op(A) * op(B)   using V_WMMA_F32_16X16X4_F32
//   - A is MxK row-major (lda); SCALED=1: A columns scaled by scaleK[k]
//   - TRANSB=0: B is KxN row-major (ldb).  TRANSB=1: memory holds Bt (NxK,
//     row-major, ldb) and the operand is Bt^T  -> NT gemm.
//   - EPI=0: C = acc ; EPI=1: C += acc ; EPI=2: C = 2*E - acc  (Newton-Schulz)
// Block: 128 threads (4 waves), 64x64 output tile, wave w owns rows w*16..+15.
//
// LDS layout: BOTH tiles are stored "fragment-major" so every WMMA operand is
// one aligned 64-bit LDS load (two consecutive k-values):
//   Asl[row][k]  row stride 18  (even stride -> 8B-aligned v2f at even k)
//   Bsl[col][k]  col stride 18  (B tile stored n-major!)
// 18*r mod 64 for r=0..15 are 16 distinct banks, word-pairs disjoint ->
// conflict-free ds_load_b64 across the 16 lanes of each half-wave.
//
// Fragment layouts per CDNA5 ISA 7.12.2:
//   A 16x4 : lane(0-15)=M, VGPR0/1 = K{0,1}; lanes 16-31: K{2,3}
//   B 4x16 : lane(0-15)=N, VGPR0/1 = K{0,1}; lanes 16-31: K{2,3}
//   C 16x16: VGPR r -> M=r (lanes 0-15) / M=r+8 (lanes 16-31), N=lane&15
// ---------------------------------------------------------------------------
#define LDSS 18   // LDS row stride in floats (even => 8B alignment holds)

template<bool TRANSB, int EPI, bool SCALED>
__global__ __launch_bounds__(128) void gemm_wmma(
    const float* __restrict__ A, int lda,
    const float* __restrict__ B, int ldb,
    float* __restrict__ C, int ldc,
    const float* __restrict__ scaleK,
    const float* __restrict__ E,
    int K)
{
    __shared__ float Asl[64 * LDSS];   // A tile: [row 0..63][k 0..15]
    __shared__ float Bsl[64 * LDSS];   // B tile: [col 0..63][k 0..15]

    const int tid  = threadIdx.x;
    const int lane = tid & 31;
    const int wave = tid >> 5;
    const int half = lane >> 4;
    const int l15  = lane & 15;

    const int rowTile = blockIdx.y * 64;
    const int colTile = blockIdx.x * 64;

    v8f acc[4];
#pragma unroll
    for (int j = 0; j < 4; ++j)
        acc[j] = (v8f){0.f, 0.f, 0.f, 0.f, 0.f, 0.f, 0.f, 0.f};

    for (int kt = 0; kt < K; kt += 16) {
        // ---- stage A tile (64 rows x 16 k), per-k scale applied at fill
        {
            const int row = tid >> 1;
            const int cb  = (tid & 1) * 8;
            const float* src = A + (size_t)(rowTile + row) * lda + kt + cb;
            float* dst = &Asl[row * LDSS + cb];
            if (SCALED) {
                const float* sc = scaleK + kt + cb;
#pragma unroll
                for (int c = 0; c < 8; ++c) dst[c] = src[c] * sc[c];
            } else {
#pragma unroll
                for (int c = 0; c < 8; ++c) dst[c] = src[c];
            }
        }
        // ---- stage B tile, stored n-major: Bsl[col][k]
        if (!TRANSB) {
            // memory B[k][n]: read coalesced along n, scatter into Bsl[n][k]
            const int kr = tid >> 3;            // k 0..15
            const int nb = (tid & 7) * 8;       // col base
            const float* src = B + (size_t)(kt + kr) * ldb + colTile + nb;
#pragma unroll
            for (int c = 0; c < 8; ++c) Bsl[(nb + c) * LDSS + kr] = src[c];
        } else {
            // memory Bt[n][k]: read coalesced along k, write contiguous
            const int nrow = tid >> 1;          // col 0..63
            const int kb   = (tid & 1) * 8;
            const float* src = B + (size_t)(colTile + nrow) * ldb + kt + kb;
            float* dst = &Bsl[nrow * LDSS + kb];
#pragma unroll
            for (int c = 0; c < 8; ++c) dst[c] = src[c];
        }
        __syncthreads();

        // ---- 4 k-steps of 4, 4 column tiles each -> 16 WMMAs per staging
#pragma unroll
        for (int kk = 0; kk < 16; kk += 4) {
            const int ak = kk + 2 * half;       // even -> 8B aligned
            const v2f a = *(const v2f*)&Asl[(wave * 16 + l15) * LDSS + ak];
#pragma unroll
            for (int j = 0; j < 4; ++j) {
                const v2f b = *(const v2f*)&Bsl[(j * 16 + l15) * LDSS + ak];
                acc[j] = __builtin_amdgcn_wmma_f32_16x16x4_f32(
                    false, a, false, b, (short)0, acc[j], false, false);
            }
        }
        __syncthreads();
    }

    // ---- epilogue
#pragma unroll
    for (int j = 0; j < 4; ++j) {
        const int col = colTile + j * 16 + l15;
#pragma unroll
        for (int r = 0; r < 8; ++r) {
            const int row = rowTile + wave * 16 + 8 * half + r;
            const size_t idx = (size_t)row * ldc + col;
            const float v = acc[j][r];
            if (EPI == 0)      C[idx] = v;
            else if (EPI == 1) C[idx] += v;
            else               C[idx] = 2.0f * E[idx] - v;
        }
    }
}

// ---------------------------------------------------------------------------
// Newton-Schulz setup: rmax = max_i sum_j |dq1[i,j]|  (symmetric => l1 == linf)
// X0 = dq1 / rmax^2
// ---------------------------------------------------------------------------
__global__ void zero_scalar(float* p) { p[0] = 0.0f; }

__global__ __launch_bounds__(256) void row_abs_max(
    const float* __restrict__ Ain, float* __restrict__ rmax)
{
    const int lane = threadIdx.x & 31;
    const int row  = blockIdx.x * 8 + (threadIdx.x >> 5);
    const float* r = Ain + (size_t)row * NDIM;
    float acc = 0.0f;
    for (int j = lane; j < NDIM; j += 32) acc += fabsf(r[j]);
#pragma unroll
    for (int off = 16; off > 0; off >>= 1) acc += __shfl_xor(acc, off);
    if (lane == 0)
        atomicMax((unsigned int*)rmax, __float_as_uint(acc)); // acc >= 0
}

__global__ __launch_bounds__(256) void ns_init(
    const float* __restrict__ Ain, const float* __restrict__ rmax,
    float* __restrict__ X)
{
    const float r   = rmax[0];
    const float inv = 1.0f / (r * r);
    const size_t i  = (size_t)blockIdx.x * 256 + threadIdx.x;
    X[i] = Ain[i] * inv;
}

// ---------------------------------------------------------------------------
extern "C" void kernel_launch(void* const* d_in, const int* in_sizes, int n_in,
                              void* d_out, int out_size, void* d_ws, size_t ws_size,
                              hipStream_t stream) {
    (void)in_sizes; (void)n_in; (void)out_size; (void)ws_size;

    const float* x  = (const float*)d_in[0];   // 2048
    const float* W1 = (const float*)d_in[1];   // 2048 x 4096
    const float* b1 = (const float*)d_in[2];   // 4096
    const float* W2 = (const float*)d_in[3];   // 4096 x 4096
    const float* b2 = (const float*)d_in[4];   // 4096
    const float* W3 = (const float*)d_in[5];   // 4096 (x1)
    float* out = (float*)d_out;                // 1024

    // workspace layout (floats), ~54.7 MB total
    float* w = (float*)d_ws;
    size_t off = 0;
    float* s1   = w + off; off += HID;
    float* h1   = w + off; off += HID;
    float* g2   = w + off; off += HID;
    float* d2v  = w + off; off += HID;
    float* g1   = w + off; off += HID;
    float* d1v  = w + off; off += HID;
    float* dq0  = w + off; off += NDIM;
    float* rhs  = w + off; off += NDIM;
    float* rmax = w + off; off += 64;
    float* Mbuf = w + off; off += (size_t)ZDIM * HID;   // M = (W1*s1) @ W2
    float* Hrt  = w + off; off += (size_t)ZDIM * NDIM;  // right 1024 cols of H
    float* Xa   = w + off; off += (size_t)NDIM * NDIM;
    float* Xb   = w + off; off += (size_t)NDIM * NDIM;
    float* Tb   = w + off; off += (size_t)NDIM * NDIM;

    float* dq2 = Hrt;                          // rows 0..1023
    float* dq1 = Hrt + (size_t)NDIM * NDIM;    // rows 1024..2047

    // 1) activations & pointwise derivative vectors
    layer1_kernel<<<HID / 256, 256, 0, stream>>>(x, W1, b1, s1, h1);
    layer2_kernel<<<HID / 256, 256, 0, stream>>>(h1, W2, b2, W3, g2, d2v);
    row_u_kernel<<<HID / 8, 256, 0, stream>>>(W2, g2, s1, g1, d1v);

    // 2) dq0 = W1[:n] @ g1
    row_matvec<<<NDIM / 8, 256, 0, stream>>>(W1, HID, HID, g1, nullptr, 1.0f,
                                             dq0, NDIM);

    // 3) M = (W1 .* s1) @ W2          (2048 x 4096, NN, scaled A)
    gemm_wmma<false, 0, true><<<dim3(HID / 64, ZDIM / 64), 128, 0, stream>>>(
        W1, HID, W2, HID, Mbuf, HID, s1, nullptr, HID);

    // 4) Hright = (W1 .* d1) @ W1botT + (M .* d2) @ MbotT   (2048 x 1024, NT)
    gemm_wmma<true, 0, true><<<dim3(NDIM / 64, ZDIM / 64), 128, 0, stream>>>(
        W1, HID, W1 + (size_t)NDIM * HID, HID, Hrt, NDIM, d1v, nullptr, HID);
    gemm_wmma<true, 1, true><<<dim3(NDIM / 64, ZDIM / 64), 128, 0, stream>>>(
        Mbuf, HID, Mbuf + (size_t)NDIM * HID, HID, Hrt, NDIM, d2v, nullptr, HID);

    // 5) rhs = dq0 - dq2 @ x[n:]
    row_matvec<<<NDIM / 8, 256, 0, stream>>>(dq2, NDIM, NDIM, x + NDIM, dq0,
                                             -1.0f, rhs, NDIM);

    // 6) Newton-Schulz inverse of dq1 (symmetric): X0 = dq1/(r*r),
    //    X <- X(2I - dq1 X), fixed 36 iterations (deterministic, graph-safe)
    zero_scalar<<<1, 1, 0, stream>>>(rmax);
    row_abs_max<<<NDIM / 8, 256, 0, stream>>>(dq1, rmax);
    ns_init<<<(NDIM * NDIM) / 256, 256, 0, stream>>>(dq1, rmax, Xa);

    float* X  = Xa;
    float* Xn = Xb;
    const dim3 gns(NDIM / 64, NDIM / 64);
    for (int it = 0; it < NS_ITERS; ++it) {
        // T = dq1 @ X   (unscaled, branch-free staging)
        gemm_wmma<false, 0, false><<<gns, 128, 0, stream>>>(
            dq1, NDIM, X, NDIM, Tb, NDIM, nullptr, nullptr, NDIM);
        // Xn = 2*X - X @ T
        gemm_wmma<false, 2, false><<<gns, 128, 0, stream>>>(
            X, NDIM, Tb, NDIM, Xn, NDIM, nullptr, X, NDIM);
        float* tmp = X; X = Xn; Xn = tmp;
    }

    // 7) out = X @ rhs
    row_matvec<<<NDIM / 8, 256, 0, stream>>>(X, NDIM, NDIM, rhs, nullptr, 1.0f,
                                             out, NDIM);
}